// RegionGaussian_48146583388833
// MI455X (gfx1250) — compile-verified
//
#include <hip/hip_runtime.h>

typedef __attribute__((ext_vector_type(2))) float v2f;
typedef __attribute__((ext_vector_type(8))) float v8f;

#define IMG_W   256
#define IMG_H   256
#define ROWS    32                   // output rows per block
#define HALO    3
#define XROWS   (ROWS + 2 * HALO)    // 38 staged rows
#define HROWS   (XROWS + 2)          // 40: +2 zero rows for K-padding to 24
#define NCH     64
#define PLANE_ELEMS (IMG_W * IMG_H)

__global__ __launch_bounds__(256) void RegionGaussian_kernel(
    const float* __restrict__ x, float* __restrict__ out) {
  extern __shared__ float smem[];
  float* sX = smem;                        // XROWS x 256 : raw x (async staged)
  float* sD = sX + XROWS * IMG_W;          // XROWS x 256 : d = 0.5*exp(x^2-x)
  float* sH = sD + XROWS * IMG_W;          // HROWS x 256 : horizontal 7-tap sums
  float* sE = sH + HROWS * IMG_W;          // ROWS  x 256 : exp(x) at center rows

  const int t     = threadIdx.x;           // 0..255
  const int bid   = blockIdx.x;            // 0..4095
  const int band  = bid & 7;               // 8 bands of 32 rows
  const int plane = bid >> 3;              // 0..511  (n*64 + ch)
  const int n     = plane >> 6;
  const int ch    = plane & 63;
  const int r0    = band * ROWS;

  const float* xp = x + (size_t)plane * PLANE_ELEMS;
  float* cp = out + (size_t)(n * 2 * NCH + ch)       * PLANE_ELEMS; // x copy
  float* rp = out + (size_t)(n * 2 * NCH + NCH + ch) * PLANE_ELEMS; // result

  const int sub  = t >> 6;                 // 0..3 : row within group-of-4
  const int col4 = (t & 63) << 2;          // float column, 16B per lane

  // ---- Phase A: async-stage x band(+halo) into sX, zero OOB halo rows ------
  for (int i = 0; i < 10; ++i) {
    const int row = i * 4 + sub;           // 0..39 (wave-uniform validity)
    if (row < XROWS) {
      const int gr = r0 - HALO + row;
      const unsigned lo = (unsigned)(size_t)&sX[row * IMG_W + col4];
      if (gr >= 0 && gr < IMG_H) {
        const float* gp = xp + gr * IMG_W + col4;
        asm volatile("global_load_async_to_lds_b128 %0, %1, off"
                     :: "v"(lo), "v"(gp) : "memory");
      } else {
        float4 z = make_float4(0.f, 0.f, 0.f, 0.f);
        *(float4*)&sX[row * IMG_W + col4] = z;   // zero-padded SAME boundary
      }
    }
  }
  asm volatile("s_wait_asynccnt 0" ::: "memory");
  __syncthreads();

  // ---- Phase B: stream x-copy half of output straight LDS -> global --------
  for (int i = 0; i < 8; ++i) {
    const int row = i * 4 + sub;           // 0..31
    const unsigned lo = (unsigned)(size_t)&sX[(HALO + row) * IMG_W + col4];
    float* gp = cp + (r0 + row) * IMG_W + col4;
    asm volatile("global_store_async_from_lds_b128 %0, %1, off"
                 :: "v"(gp), "v"(lo) : "memory");
  }

  // ---- Phase C: d = 0.5*exp(x^2-x) and exp(x) at center rows ---------------
  for (int i = 0; i < XROWS; ++i) {
    const int gr = r0 - HALO + i;          // wave-uniform
    const float xv = sX[i * IMG_W + t];
    sD[i * IMG_W + t] = (gr >= 0 && gr < IMG_H)
                          ? 0.5f * __expf(xv * xv - xv) : 0.0f;
    if (gr >= r0 && gr < r0 + ROWS)
      sE[(gr - r0) * IMG_W + t] = __expf(xv);
  }
  __syncthreads();

  // ---- Phase D: horizontal 7-tap zero-padded sum, D -> H -------------------
  for (int i = 0; i < XROWS; ++i) {
    float s = 0.0f;
#pragma unroll
    for (int dc = -3; dc <= 3; ++dc) {
      const int c = t + dc;
      if (c >= 0 && c < IMG_W) s += sD[i * IMG_W + c];
    }
    sH[i * IMG_W + t] = s;
  }
  sH[(XROWS + 0) * IMG_W + t] = 0.0f;      // K-padding rows (K = 24)
  sH[(XROWS + 1) * IMG_W + t] = 0.0f;
  __syncthreads();

  // ---- Phase E: vertical 7-tap sum as banded WMMA, normalize, store --------
  const int wave  = t >> 5;
  const int lane  = t & 31;
  const int ln16  = lane & 15;             // A: M row / B: N col
  const int khalf = (lane >> 4) << 1;      // K sub-index per WMMA VGPR layout
  const int mbase = (lane >> 4) << 3;      // C/D: lanes 16-31 hold M+8

  for (int q = 0; q < 4; ++q) {
    const int tile  = wave + (q << 3);     // 32 tiles: 2 row-tiles x 16 col-tiles
    const int rt    = tile >> 4;
    const int col0  = (tile & 15) << 4;
    const int hbase = rt << 4;             // local H-row base for this row-tile

    v8f acc = {};
#pragma unroll
    for (int kk = 0; kk < 6; ++kk) {
      const int k0 = kk * 4 + khalf;
      v2f a, b;
      // band matrix: W[m][k] = 1 iff 0 <= k - m <= 6
      a.x = ((unsigned)(k0     - ln16) <= 6u) ? 1.0f : 0.0f;
      a.y = ((unsigned)(k0 + 1 - ln16) <= 6u) ? 1.0f : 0.0f;
      b.x = sH[(hbase + k0    ) * IMG_W + col0 + ln16];
      b.y = sH[(hbase + k0 + 1) * IMG_W + col0 + ln16];
      acc = __builtin_amdgcn_wmma_f32_16x16x4_f32(
          false, a, false, b, (short)0, acc, false, false);
    }

#pragma unroll
    for (int v = 0; v < 8; ++v) {
      const int M    = v + mbase;
      const int orow = r0 + hbase + M;
      const int ocol = col0 + ln16;
      int lov = orow - 3; if (lov < 0) lov = 0;
      int hiv = orow + 3; if (hiv > IMG_H - 1) hiv = IMG_H - 1;
      int loh = ocol - 3; if (loh < 0) loh = 0;
      int hih = ocol + 3; if (hih > IMG_W - 1) hih = IMG_W - 1;
      const float cnt = (float)((hiv - lov + 1) * (hih - loh + 1));
      const float res = acc[v] * __builtin_amdgcn_rcpf(cnt)
                        * sE[(hbase + M) * IMG_W + ocol];
      rp[orow * IMG_W + ocol] = res;
    }
  }
  // async copy-out stores complete before S_ENDPGM (implicit wait-idle),
  // but wait explicitly to be safe w.r.t. counter semantics
  asm volatile("s_wait_asynccnt 0" ::: "memory");
}

extern "C" void kernel_launch(void* const* d_in, const int* in_sizes, int n_in,
                              void* d_out, int out_size, void* d_ws, size_t ws_size,
                              hipStream_t stream) {
  (void)in_sizes; (void)n_in; (void)out_size; (void)d_ws; (void)ws_size;
  const float* x = (const float*)d_in[0];
  float* out = (float*)d_out;
  const size_t smem =
      (size_t)(XROWS + XROWS + HROWS + ROWS) * IMG_W * sizeof(float); // 151,552 B
  const int blocks = 512 * (IMG_H / ROWS);  // 4096
  RegionGaussian_kernel<<<dim3(blocks), dim3(256), smem, stream>>>(x, out);
}